// Encoder_1468878815894
// MI455X (gfx1250) — compile-verified
//
#include <hip/hip_runtime.h>

typedef __attribute__((ext_vector_type(2))) float v2f;
typedef __attribute__((ext_vector_type(8))) float v8f;

#define N_NODES 10000
#define N_EDGES 320000
#define DIM_D   128
#define DIM_H   256

// ---------------------------------------------------------------------------
// Zero-fill (float4 streaming stores) for the adjacency region.
// ---------------------------------------------------------------------------
__global__ __launch_bounds__(256) void zero_f4(float4* __restrict__ p, long n4) {
    long i = (long)blockIdx.x * blockDim.x + threadIdx.x;
    long stride = (long)gridDim.x * blockDim.x;
    float4 zv; zv.x = 0.f; zv.y = 0.f; zv.z = 0.f; zv.w = 0.f;
    for (; i < n4; i += stride) p[i] = zv;
}

// ---------------------------------------------------------------------------
// Scatter-add edges into dense adjacency (coalesces duplicate edges like the
// reference's .at[].add). global_atomic_add_f32.
// ---------------------------------------------------------------------------
__global__ __launch_bounds__(256) void scatter_edges(const int* __restrict__ ei,
                                                     const float* __restrict__ ew,
                                                     float* __restrict__ adj,
                                                     int E, int N) {
    int e = blockIdx.x * blockDim.x + threadIdx.x;
    if (e < E) {
        int s = ei[e];
        int d = ei[E + e];
        atomicAdd(adj + (long)s * N + d, ew[e]);
    }
}

// ---------------------------------------------------------------------------
// Per-row sum -> store reciprocal 1/(rowsum+eps). One block per row, float4
// loads, LDS tree reduction.
// ---------------------------------------------------------------------------
__global__ __launch_bounds__(256) void rowsum_inv(const float* __restrict__ adj,
                                                  float* __restrict__ inv, int N) {
    __shared__ float red[256];
    const long base = (long)blockIdx.x * N;
    const float4* p = (const float4*)(adj + base);
    const int n4 = N >> 2;  // 2500
    float s = 0.f;
    for (int i = threadIdx.x; i < n4; i += 256) {
        float4 v = p[i];
        s += v.x + v.y + v.z + v.w;
    }
    red[threadIdx.x] = s;
    __syncthreads();
    for (int off = 128; off > 0; off >>= 1) {
        if ((int)threadIdx.x < off) red[threadIdx.x] += red[threadIdx.x + off];
        __syncthreads();
    }
    if (threadIdx.x == 0) inv[blockIdx.x] = 1.0f / (red[0] + 1e-20f);
}

// ---------------------------------------------------------------------------
// rw[i,j] = adj[i,j] * inv[j]  (reference's broadcast over the LAST axis).
// One read pass, two write streams (rw1 in place, rw2 fused).
// grid = (ceil((N/4)/256), N)
// ---------------------------------------------------------------------------
__global__ __launch_bounds__(256) void normalize_rw(const float* __restrict__ inv,
                                                    float* __restrict__ rw1,
                                                    float* __restrict__ rw2, int N) {
    int j4 = blockIdx.x * 256 + threadIdx.x;   // float4 index within row
    int n4 = N >> 2;                           // 2500
    if (j4 >= n4) return;
    long idx = (long)blockIdx.y * N + (long)j4 * 4;
    float4 a = *(const float4*)(rw1 + idx);
    float4 w = *(const float4*)(inv + (long)j4 * 4);
    a.x *= w.x; a.y *= w.y; a.z *= w.z; a.w *= w.w;
    *(float4*)(rw1 + idx) = a;
    *(float4*)(rw2 + idx) = a;
}

// ---------------------------------------------------------------------------
// z = x @ W_enc via V_WMMA_F32_16X16X4_F32 (native f32 matrix pipe).
// One wave32 per 16x16 output tile; 8 waves per block; K=128 as 32 x (K=4).
//
// Fragment layouts (ISA 7.12.2, 32-bit data):
//   A 16x4 : lanes 0-15 -> M=lane,    a[0..1] = K=0,1
//            lanes16-31 -> M=lane-16, a[0..1] = K=2,3
//   B 4x16 : lanes 0-15 -> N=lane,    b[0..1] = K=0,1
//            lanes16-31 -> N=lane-16, b[0..1] = K=2,3
//   C/D    : VGPR r -> (M=r, N=lane) lanes 0-15 ; (M=8+r, N=lane-16) lanes 16-31
// Epilogue writes each value to z, z1, z2 (reference duplicates the encoding).
// ---------------------------------------------------------------------------
__global__ __launch_bounds__(256) void gemm_wmma_f32(const float* __restrict__ x,
                                                     const float* __restrict__ W,
                                                     float* __restrict__ z,
                                                     float* __restrict__ z1,
                                                     float* __restrict__ z2) {
    const int lane  = threadIdx.x & 31;
    const int wave  = threadIdx.x >> 5;
    const int tile  = blockIdx.x * 8 + wave;  // 0..9999, exact
    const int tm    = tile >> 4;              // 0..624 (M tiles)
    const int tn    = tile & 15;              // 0..15  (N tiles)
    const int l15   = lane & 15;
    const int khalf = lane >> 4;              // 0: K pair {0,1}; 1: K pair {2,3}

    const int arow = tm * 16 + l15;           // A row this lane feeds
    const int col  = tn * 16 + l15;           // B/C/D column this lane feeds

    v8f acc = {};
    const float* __restrict__ ap = x + (long)arow * DIM_D + khalf * 2;
    const float* __restrict__ bp = W + (long)(khalf * 2) * DIM_H + col;
#pragma unroll 4
    for (int k = 0; k < DIM_D; k += 4) {
        v2f a, b;
        a.x = ap[k];
        a.y = ap[k + 1];
        b.x = bp[(long)k * DIM_H];
        b.y = bp[(long)(k + 1) * DIM_H];
        acc = __builtin_amdgcn_wmma_f32_16x16x4_f32(
            /*neg_a=*/false, a, /*neg_b=*/false, b,
            /*c_mod=*/(short)0, acc, /*reuse_a=*/false, /*reuse_b=*/false);
    }

    const int mbase = tm * 16 + khalf * 8;
#pragma unroll
    for (int r = 0; r < 8; ++r) {
        const long o = (long)(mbase + r) * DIM_H + col;
        const float v = acc[r];
        z[o]  = v;
        z1[o] = v;
        z2[o] = v;
    }
}

// ---------------------------------------------------------------------------
// Orchestration. Output layout (flat, f32): z | z1 | z2 | rw1 | rw2
// ---------------------------------------------------------------------------
extern "C" void kernel_launch(void* const* d_in, const int* in_sizes, int n_in,
                              void* d_out, int out_size, void* d_ws, size_t ws_size,
                              hipStream_t stream) {
    const float* x  = (const float*)d_in[0];   // (N, D)
    const int*   ei = (const int*)  d_in[1];   // (2, E)
    const float* ew = (const float*)d_in[2];   // (E,)
    const float* W  = (const float*)d_in[3];   // (D, H)

    const int N = N_NODES, E = N_EDGES, H = DIM_H;

    float* out = (float*)d_out;
    float* z   = out;
    float* z1  = out + (long)N * H;
    float* z2  = out + 2L * (long)N * H;
    float* rw1 = out + 3L * (long)N * H;
    float* rw2 = rw1 + (long)N * N;

    float* inv = (float*)d_ws;                 // N floats of scratch

    // 1) adj := 0 (built in place inside the rw1 output region)
    zero_f4<<<4096, 256, 0, stream>>>((float4*)rw1, (long)N * N / 4);

    // 2) adj[s,d] += w  (duplicate-edge coalescing via f32 atomics)
    scatter_edges<<<(E + 255) / 256, 256, 0, stream>>>(ei, ew, rw1, E, N);

    // 3) inv[i] = 1/(sum_j adj[i,j] + 1e-20)
    rowsum_inv<<<N, 256, 0, stream>>>(rw1, inv, N);

    // 4) rw1[i,j] = adj[i,j]*inv[j]; rw2 = rw1 (fused single read pass)
    {
        dim3 grid((N / 4 + 255) / 256, N);     // (10, 10000)
        normalize_rw<<<grid, 256, 0, stream>>>(inv, rw1, rw2, N);
    }

    // 5) z = x @ W_enc (f32 WMMA), duplicated into z1, z2
    gemm_wmma_f32<<<(N / 16) * (H / 16) / 8, 256, 0, stream>>>(x, W, z, z1, z2);
}